// MergingModel_76244259438723
// MI455X (gfx1250) — compile-verified
//
#include <hip/hip_runtime.h>
#include <math.h>

// ---------------- problem constants (match reference) ----------------
#define N_OBS   2000000
#define N_REFL  250000
#define N_IMG   2000
#define MC      32
#define D_META  6
#define D_EMB   32
#define D_IN    44      // 2 + 2 + 6 + 2 + 32
#define HIDDEN  64
#define LOG2PI_F 1.8378770664093453f

// CDNA5 WMMA fragment types (fp32 16x16x4: A/B = 2 VGPRs, C/D = 8 VGPRs, wave32)
typedef __attribute__((ext_vector_type(2))) float v2f;
typedef __attribute__((ext_vector_type(8))) float v8f;

__device__ __forceinline__ float softplus_f(float x) {
    return fmaxf(x, 0.f) + log1pf(expf(-fabsf(x)));
}

// ---------------- pass 0: zero accumulators (d_ws is poisoned 0xAA) ----------------
__global__ void init_kernel(float* ll_ws, double* acc) {
    int i = blockIdx.x * 256 + threadIdx.x;
    if (i < N_IMG * 2) ll_ws[i] = 0.f;
    if (i < 8) acc[i] = 0.0;
}

// ---------------- pass 1: per-reflection sq, KL, zT = (q_loc + sq*zeps)^T ----------------
__global__ __launch_bounds__(256) void refl_kernel(const float* __restrict__ q_loc,
                                                   const float* __restrict__ q_scale_raw,
                                                   const float* __restrict__ zeps,
                                                   float* __restrict__ sq_ws,
                                                   float* __restrict__ zT,
                                                   double* __restrict__ acc) {
    int r = blockIdx.x * 256 + threadIdx.x;
    float kl = 0.f;
    if (r < N_REFL) {
        float sq = softplus_f(q_scale_raw[r]);
        float ql = q_loc[r];
        sq_ws[r] = sq;
        kl = -logf(sq) + 0.5f * (sq * sq + ql * ql) - 0.5f;
#pragma unroll
        for (int m = 0; m < MC; ++m)   // zeps is [MC, N_REFL]; read coalesced per m
            zT[(size_t)r * MC + m] = ql + sq * zeps[(size_t)m * N_REFL + r];
    }
    for (int off = 16; off > 0; off >>= 1) kl += __shfl_xor(kl, off, 32);
    if ((threadIdx.x & 31) == 0) atomicAdd(&acc[0], (double)kl);   // global_atomic_add_f64
}

// ---------------- pass 2: hot MLP + MC likelihood (WMMA fp32) ----------------
// block = 128 threads = 4 waves; each wave owns a 16-observation tile.
// W1 B-fragments and op-invariant A-fragments are hoisted into registers so the
// second op's GEMM runs entirely out of VGPRs (no repeated LDS fragment fetches).
__global__ __launch_bounds__(128) void mlp_kernel(
    const float* __restrict__ I, const float* __restrict__ SigI,
    const int* __restrict__ image_id, const float* __restrict__ metadata,
    const float* __restrict__ wavelength, const float* __restrict__ dHKL,
    const int* __restrict__ refl_ids, const float* __restrict__ q_loc,
    const float* __restrict__ sq_ws, const float* __restrict__ img_emb,
    const float* __restrict__ W1, const float* __restrict__ b1,
    const float* __restrict__ W2, const float* __restrict__ b2,
    const float* __restrict__ zT, const float* __restrict__ seps,
    float* __restrict__ ll_ws, float* __restrict__ Ipred_ws) {

    __shared__ float sW1[D_IN * 65];     // padded rows, 44x64 weights
    __shared__ float sB1[HIDDEN];
    __shared__ float sW2[HIDDEN * 2];
    __shared__ float sB2[2];
    __shared__ float sF[4][16][45];      // per-wave feature tile [obs][k], pad 45
    __shared__ float sH[4][16][65];      // per-wave hidden tile [obs][h], pad 65
    __shared__ float sLoc[4][16], sLs[4][16];
    __shared__ int   sImg[4][16], sRid[4][16];

    const int lane  = threadIdx.x & 31;
    const int w     = threadIdx.x >> 5;           // wave in block: 0..3
    const int l16   = lane & 15;
    const int half  = lane >> 4;                  // 0/1
    const int klo   = half * 2;                   // A/B fragment K low index per ISA layout
    const int obase = blockIdx.x * 64 + w * 16;   // 31250 blocks * 64 obs = 2,000,000 exactly

    // stage weights (shared by whole block)
    for (int i = threadIdx.x; i < D_IN * HIDDEN; i += 128)
        sW1[(i / HIDDEN) * 65 + (i % HIDDEN)] = W1[i];
    for (int i = threadIdx.x; i < HIDDEN; i += 128) sB1[i] = b1[i];
    for (int i = threadIdx.x; i < HIDDEN * 2; i += 128) sW2[i] = W2[i];
    if (threadIdx.x < 2) sB2[threadIdx.x] = b2[threadIdx.x];

    if (lane < 16) sImg[w][lane] = image_id[obase + lane];
    __syncthreads();

    // static feature columns (identical for both ops): j = 2..43
    for (int idx = lane; idx < 16 * D_IN; idx += 32) {
        int o = idx / D_IN, j = idx % D_IN;
        if (j < 2) continue;
        int obs = obase + o;
        float v;
        if (j == 2)       v = I[obs];
        else if (j == 3)  v = SigI[obs];
        else if (j < 10)  v = metadata[obs * D_META + (j - 4)];
        else if (j == 10) v = wavelength[obs];
        else if (j == 11) { float d = dHKL[obs]; v = 1.f / (d * d); }
        else              v = img_emb[sImg[w][o] * D_EMB + (j - 12)];
        sF[w][o][j] = v;
    }
    __syncthreads();

    // ---- hoist op-invariant WMMA fragments into registers ----
    // B (W1): 11 K-steps x 4 N-tiles; per ISA layout B row K striped across lanes,
    // lanes 0-15 -> K={k0,k0+1}, lanes 16-31 -> K={k0+2,k0+3}, col = nt*16+l16.
    v2f bfr[11][4];
#pragma unroll
    for (int kk = 0; kk < 11; ++kk) {
        int k0 = 4 * kk + klo;
#pragma unroll
        for (int nt = 0; nt < 4; ++nt) {
            bfr[kk][nt].x = sW1[k0 * 65 + nt * 16 + l16];
            bfr[kk][nt].y = sW1[(k0 + 1) * 65 + nt * 16 + l16];
        }
    }
    // A (features): K-steps 1..10 read only static columns (4..43) -> op-invariant.
    v2f afr[11];
#pragma unroll
    for (int kk = 1; kk < 11; ++kk) {
        int k0 = 4 * kk + klo;
        afr[kk].x = sF[w][l16][k0];
        afr[kk].y = sF[w][l16][k0 + 1];
    }

    for (int op = 0; op < 2; ++op) {
        __syncthreads();   // previous-iteration readers done before cols 0/1 + sRid rewrite
        if (lane < 16) {
            int o = lane, obs = obase + o;
            int rid = refl_ids[op * N_OBS + obs];
            sRid[w][o] = rid;
            sF[w][o][0] = q_loc[rid];   // Imodel = [q_loc, sq]
            sF[w][o][1] = sq_ws[rid];
        }
        __syncthreads();

        // only the kk=0 A-fragment touches per-op columns 0/1
        afr[0].x = sF[w][l16][klo];
        afr[0].y = sF[w][l16][klo + 1];

        // ---- GEMM1: feats[16,44] @ W1[44,64], fp32 WMMA, all-register operands ----
        v8f acc[4];
        const v8f vzero = {0.f,0.f,0.f,0.f,0.f,0.f,0.f,0.f};
#pragma unroll
        for (int nt = 0; nt < 4; ++nt) acc[nt] = vzero;
#pragma unroll
        for (int kk = 0; kk < 11; ++kk) {
#pragma unroll
            for (int nt = 0; nt < 4; ++nt) {
                acc[nt] = __builtin_amdgcn_wmma_f32_16x16x4_f32(
                    false, afr[kk], false, bfr[kk][nt], (short)0, acc[nt], false, false);
            }
        }

        // bias + relu, scatter C layout (row m = r + 8*half, col n = nt*16+l16) to LDS
#pragma unroll
        for (int nt = 0; nt < 4; ++nt) {
            float bb = sB1[nt * 16 + l16];
#pragma unroll
            for (int r = 0; r < 8; ++r)
                sH[w][r + 8 * half][nt * 16 + l16] = fmaxf(acc[nt][r] + bb, 0.f);
        }
        __syncthreads();

        // ---- GEMM2: h[16,64] @ W2[64,2] : 32 lanes = 16 obs x 2 cols ----
        {
            int o = lane >> 1, col = lane & 1;
            float s = sB2[col];
#pragma unroll 8
            for (int k = 0; k < HIDDEN; ++k) s += sH[w][o][k] * sW2[k * 2 + col];
            if (col == 0) sLoc[w][o] = s; else sLs[w][o] = s;
        }
        __syncthreads();

        // ---- MC phase: lane = mc sample; loop over the wave's 16 observations ----
        for (int o = 0; o < 16; ++o) {
            int obs = obase + o;
            float loc = sLoc[w][o];
            float sp  = softplus_f(sLs[w][o]);
            float se  = seps[(size_t)obs * MC + lane];          // coalesced 128B
            float zv  = zT[(size_t)sRid[w][o] * MC + lane];     // L2-resident gather
            float ip  = zv * (loc + sp * se);
            float Iv  = sF[w][o][2], Sv = sF[w][o][3];
            float t   = (ip - Iv) / Sv;
            float llv = -0.5f * t * t - logf(Sv) - 0.5f * LOG2PI_F;
            float sll = llv, sip = ip;
            for (int off = 16; off > 0; off >>= 1) {
                sll += __shfl_xor(sll, off, 32);
                sip += __shfl_xor(sip, off, 32);
            }
            if (lane == 0) {
                atomicAdd(&ll_ws[sImg[w][o] * 2 + op], sll);
                Ipred_ws[(size_t)obs * 2 + op] = sip * (1.f / MC);
            }
        }
    }
}

// ---------------- pass 3: per-image mean over mc, max/argmax over ops ----------------
__global__ void argmax_kernel(const float* __restrict__ ll_ws, int* __restrict__ opidx,
                              double* __restrict__ acc) {
    int i = blockIdx.x * 256 + threadIdx.x;
    float lmax = 0.f;
    if (i < N_IMG) {
        float l0 = ll_ws[i * 2 + 0] * (1.f / MC);
        float l1 = ll_ws[i * 2 + 1] * (1.f / MC);
        opidx[i] = (l1 > l0) ? 1 : 0;     // argmax: first index wins ties
        lmax = fmaxf(l0, l1);
    }
    for (int off = 16; off > 0; off >>= 1) lmax += __shfl_xor(lmax, off, 32);
    if ((threadIdx.x & 31) == 0) atomicAdd(&acc[1], (double)lmax);
}

// ---------------- pass 4: weighted-Pearson raw moments ----------------
__global__ __launch_bounds__(256) void pearson_kernel(const float* __restrict__ I,
                                                      const float* __restrict__ SigI,
                                                      const int* __restrict__ image_id,
                                                      const float* __restrict__ Ipred_ws,
                                                      const int* __restrict__ opidx,
                                                      double* __restrict__ acc) {
    int n = blockIdx.x * 256 + threadIdx.x;
    float wv = 0.f, x = 0.f, y = 0.f;
    if (n < N_OBS) {
        x = I[n];
        float s = SigI[n];
        wv = 1.f / (s * s);
        y = Ipred_ws[(size_t)n * 2 + opidx[image_id[n]]];
    }
    float m[6] = { wv, wv * x, wv * y, wv * x * x, wv * y * y, wv * x * y };
#pragma unroll
    for (int j = 0; j < 6; ++j)
        for (int off = 16; off > 0; off >>= 1) m[j] += __shfl_xor(m[j], off, 32);
    if ((threadIdx.x & 31) == 0) {
#pragma unroll
        for (int j = 0; j < 6; ++j) atomicAdd(&acc[2 + j], (double)m[j]);
    }
}

// ---------------- pass 5: finalize elbo + cc ----------------
__global__ void finalize_kernel(const double* __restrict__ acc, float* __restrict__ out) {
    double kl   = acc[0] / (double)N_REFL;
    double elbo = -(acc[1] / (double)N_IMG) + kl;       // KL_WEIGHT = 1
    double sw = acc[2], swx = acc[3], swy = acc[4], swxx = acc[5], swyy = acc[6], swxy = acc[7];
    double z = 1.0 / sw;
    double mx = z * swx, my = z * swy;
    double cxy = z * swxy - mx * my;
    double cx  = z * swxx - mx * mx;
    double cy  = z * swyy - my * my;
    out[0] = (float)elbo;
    out[1] = (float)(cxy / sqrt(cx * cy));
}

// ---------------- host side ----------------
extern "C" void kernel_launch(void* const* d_in, const int* in_sizes, int n_in,
                              void* d_out, int out_size, void* d_ws, size_t ws_size,
                              hipStream_t stream) {
    const float* I          = (const float*)d_in[0];
    const float* SigI       = (const float*)d_in[1];
    const int*   image_id   = (const int*)  d_in[2];
    const float* metadata   = (const float*)d_in[3];
    const float* wavelength = (const float*)d_in[4];
    const float* dHKL       = (const float*)d_in[5];
    const int*   refl_ids   = (const int*)  d_in[6];
    const float* q_loc      = (const float*)d_in[7];
    const float* q_scale    = (const float*)d_in[8];
    const float* img_emb    = (const float*)d_in[9];
    const float* W1         = (const float*)d_in[10];
    const float* b1         = (const float*)d_in[11];
    const float* W2         = (const float*)d_in[12];
    const float* b2         = (const float*)d_in[13];
    const float* zeps       = (const float*)d_in[14];
    const float* seps       = (const float*)d_in[15];
    float* out              = (float*)d_out;

    // workspace layout (~52 MB): sq | zT | Ipred | ll | opidx | accums
    char* ws = (char*)d_ws;
    float*  sq_ws    = (float*) (ws);                                     // 1.0 MB
    float*  zT       = (float*) (ws + (size_t)(1 << 20));                 // 32.0 MB
    float*  Ipred_ws = (float*) (ws + (size_t)(1 << 20) + ((size_t)32 << 20)); // 16.0 MB
    size_t  llOff    = (size_t)(1 << 20) + ((size_t)32 << 20) + ((size_t)16 << 20);
    float*  ll_ws    = (float*) (ws + llOff);                             // 16 KB
    int*    opidx    = (int*)   (ws + llOff + (1 << 16));                 // 8 KB
    double* acc      = (double*)(ws + llOff + (2 << 16));                 // 64 B

    init_kernel<<<16, 256, 0, stream>>>(ll_ws, acc);
    refl_kernel<<<(N_REFL + 255) / 256, 256, 0, stream>>>(q_loc, q_scale, zeps, sq_ws, zT, acc);
    mlp_kernel<<<N_OBS / 64, 128, 0, stream>>>(I, SigI, image_id, metadata, wavelength, dHKL,
                                               refl_ids, q_loc, sq_ws, img_emb,
                                               W1, b1, W2, b2, zT, seps, ll_ws, Ipred_ws);
    argmax_kernel<<<(N_IMG + 255) / 256, 256, 0, stream>>>(ll_ws, opidx, acc);
    pearson_kernel<<<(N_OBS + 255) / 256, 256, 0, stream>>>(I, SigI, image_id, Ipred_ws, opidx, acc);
    finalize_kernel<<<1, 1, 0, stream>>>(acc, out);
}